// PatchcoreModel_5540507812093
// MI455X (gfx1250) — compile-verified
//
#include <hip/hip_runtime.h>

typedef __attribute__((ext_vector_type(2))) float v2f;
typedef __attribute__((ext_vector_type(8))) float v8f;

#define C_DIM   384
#define KNN     9
#define BATCH   8
#define PH      28
#define PW      28
#define OUT_HW  224
#define KSIZE   33
#define KRAD    16
#define A_STR   388   // 384 + 4 pad: row-stride %64 == 4 -> conflict-free A frags
#define T_STR   68    // 64 + 4 pad for tile buffer

__device__ __forceinline__ void topk_insert(float (&s)[KNN], float v) {
    if (v < s[KNN - 1]) {
        s[KNN - 1] = v;
#pragma unroll
        for (int i = KNN - 1; i > 0; --i) {
            float a = s[i - 1];
            if (s[i] < a) { s[i - 1] = s[i]; s[i] = a; }
        }
    }
}

// ---------------------------------------------------------------- ||y||^2
__global__ void __launch_bounds__(256) ynorm_kernel(const float* __restrict__ mem,
                                                    float* __restrict__ yn, int M) {
    const int lane = threadIdx.x & 31;
    const int wave = (blockIdx.x * blockDim.x + threadIdx.x) >> 5;
    const int nwaves = (gridDim.x * blockDim.x) >> 5;
    for (int row = wave; row < M; row += nwaves) {
        const float* p = mem + (size_t)row * C_DIM + lane * 4;
        float s = 0.f;
#pragma unroll
        for (int i = 0; i < 3; ++i) {
            float4 v = *(const float4*)(p + i * 128);
            s += v.x * v.x + v.y * v.y + v.z * v.z + v.w * v.w;
        }
#pragma unroll
        for (int off = 16; off; off >>= 1) s += __shfl_xor(s, off, 32);
        if (lane == 0) yn[row] = s;
    }
}

// ------------------------------------------- fused cdist + top-9 (the hot loop)
__global__ void __launch_bounds__(256) cdist_topk_kernel(
        const float* __restrict__ emb, const float* __restrict__ mem,
        const float* __restrict__ yn, float* __restrict__ pscore, int M) {
    __shared__ float As[16 * A_STR];
    __shared__ float Tile[8 * 16 * T_STR];
    __shared__ float Lists[256 * KNN];
    __shared__ float Xn[16];

    const int tid  = threadIdx.x;
    const int wave = tid >> 5;
    const int lane = tid & 31;
    const int rowBlock = blockIdx.x * 16;

    // stage 16 embedding rows into LDS (coalesced)
    for (int i = tid; i < 16 * C_DIM; i += 256) {
        int r = i / C_DIM, c = i - r * C_DIM;
        As[r * A_STR + c] = emb[(size_t)(rowBlock + r) * C_DIM + c];
    }
    __syncthreads();
    if (tid < 16) {
        float s = 0.f;
        for (int c = 0; c < C_DIM; ++c) { float v = As[tid * A_STR + c]; s += v * v; }
        Xn[tid] = s;
    }
    __syncthreads();

    const int r    = lane & 15;
    const int half = lane >> 4;
    const int c0   = half * 2;          // K-pair offset per WMMA f32 A/B layout

    float s[KNN];
#pragma unroll
    for (int i = 0; i < KNN; ++i) s[i] = 3.0e38f;

    float* tb = &Tile[wave * 16 * T_STR];

    for (int mBase = wave * 64; mBase < M; mBase += 8 * 64) {
        const float* bp[4];
        float ynv[4];
#pragma unroll
        for (int t = 0; t < 4; ++t) {
            int col = mBase + t * 16 + r;
            if (col > M - 1) col = M - 1;           // clamp (masked at insert)
            bp[t]  = mem + (size_t)col * C_DIM + c0;
            ynv[t] = yn[col];
        }
        v8f acc0 = {}, acc1 = {}, acc2 = {}, acc3 = {};

#pragma unroll 4
        for (int k = 0; k < C_DIM; k += 4) {
            v2f a  = *(const v2f*)&As[r * A_STR + k + c0];
            v2f b0 = *(const v2f*)(bp[0] + k);
            v2f b1 = *(const v2f*)(bp[1] + k);
            v2f b2 = *(const v2f*)(bp[2] + k);
            v2f b3 = *(const v2f*)(bp[3] + k);
            acc0 = __builtin_amdgcn_wmma_f32_16x16x4_f32(false, a, false, b0, (short)0, acc0, false, false);
            acc1 = __builtin_amdgcn_wmma_f32_16x16x4_f32(false, a, false, b1, (short)0, acc1, false, false);
            acc2 = __builtin_amdgcn_wmma_f32_16x16x4_f32(false, a, false, b2, (short)0, acc2, false, false);
            acc3 = __builtin_amdgcn_wmma_f32_16x16x4_f32(false, a, false, b3, (short)0, acc3, false, false);
        }

        // fold norms, transpose via LDS:  D layout = VGPR g -> row g|g+8, lane&15 -> col
#pragma unroll
        for (int g = 0; g < 8; ++g) {
            int row = g + half * 8;
            float xn = Xn[row];
            tb[row * T_STR + r +  0] = xn + ynv[0] - 2.f * acc0[g];
            tb[row * T_STR + r + 16] = xn + ynv[1] - 2.f * acc1[g];
            tb[row * T_STR + r + 32] = xn + ynv[2] - 2.f * acc2[g];
            tb[row * T_STR + r + 48] = xn + ynv[3] - 2.f * acc3[g];
        }
        asm volatile("s_wait_dscnt 0x0" ::: "memory");   // intra-wave LDS RAW

        // each lane scans one row / 32 columns, keeps running top-9
#pragma unroll
        for (int u = 0; u < 8; ++u) {
            float4 v = *(const float4*)&tb[r * T_STR + half * 32 + u * 4];
            int cb = mBase + half * 32 + u * 4;
            if (cb + 0 < M) topk_insert(s, v.x);
            if (cb + 1 < M) topk_insert(s, v.y);
            if (cb + 2 < M) topk_insert(s, v.z);
            if (cb + 3 < M) topk_insert(s, v.w);
        }
    }

#pragma unroll
    for (int i = 0; i < KNN; ++i) Lists[tid * KNN + i] = s[i];
    __syncthreads();

    if (tid < 16) {                       // merge 16 partial lists per row
        float f[KNN];
#pragma unroll
        for (int i = 0; i < KNN; ++i) f[i] = 3.0e38f;
        for (int w = 0; w < 8; ++w)
            for (int hh = 0; hh < 2; ++hh) {
                const float* Lp = &Lists[(w * 32 + hh * 16 + tid) * KNN];
                for (int i = 0; i < KNN; ++i) topk_insert(f, Lp[i]);
            }
        for (int i = 0; i < KNN; ++i)
            pscore[(size_t)(rowBlock + tid) * KNN + i] = sqrtf(fmaxf(f[i], 0.f));
    }
}

// ------------------------------------------------------- anomaly score (per batch)
__global__ void __launch_bounds__(256) score_kernel(const float* __restrict__ ps,
                                                    float* __restrict__ outScore) {
    __shared__ float sv[256];
    __shared__ int   si[256];
    const int b = blockIdx.x, tid = threadIdx.x, hw = PH * PW;
    float best = -3.0e38f; int bidx = 0;
    for (int p = tid; p < hw; p += 256) {
        float v = ps[(size_t)(b * hw + p) * KNN];
        if (v > best) { best = v; bidx = p; }
    }
    sv[tid] = best; si[tid] = bidx; __syncthreads();
    for (int off = 128; off; off >>= 1) {
        if (tid < off) {
            float v = sv[tid + off]; int ix = si[tid + off];
            if (v > sv[tid] || (v == sv[tid] && ix < si[tid])) { sv[tid] = v; si[tid] = ix; }
        }
        __syncthreads();
    }
    if (tid == 0) {
        int amax = si[0]; float mx = sv[0];
        float c[KNN], m = -3.0e38f;
        for (int j = 0; j < KNN; ++j) { c[j] = ps[(size_t)(b * hw + amax) * KNN + j]; m = fmaxf(m, c[j]); }
        float sum = 0.f;
        for (int j = 0; j < KNN; ++j) sum += expf(c[j] - m);
        outScore[b] = (1.f - 1.f / sum) * mx;
    }
}

// ------------------------------------------------------- Gaussian weights (33 taps)
__global__ void gweights_kernel(float* __restrict__ g) {
    if (threadIdx.x == 0 && blockIdx.x == 0) {
        float tmp[KSIZE], sum = 0.f;
        for (int t = 0; t < KSIZE; ++t) {
            float d = (float)(t - KRAD);
            tmp[t] = expf(-(d * d) / 32.0f);   // 2*sigma^2, sigma=4
            sum += tmp[t];
        }
        for (int t = 0; t < KSIZE; ++t) g[t] = tmp[t] / sum;
    }
}

// ------------------------------------------------------- 28 -> 224 bilinear (half-pixel)
__global__ void __launch_bounds__(256) upsample_kernel(const float* __restrict__ ps,
                                                       float* __restrict__ up) {
    int idx = blockIdx.x * blockDim.x + threadIdx.x;
    int j = idx % OUT_HW, rest = idx / OUT_HW;
    int i = rest % OUT_HW, b = rest / OUT_HW;
    float sy = (i + 0.5f) * (1.0f / 8.0f) - 0.5f;
    float sx = (j + 0.5f) * (1.0f / 8.0f) - 0.5f;
    int y0 = (int)floorf(sy), x0 = (int)floorf(sx);
    float fy = sy - y0, fx = sx - x0;
    int y1 = min(max(y0 + 1, 0), PH - 1), x1 = min(max(x0 + 1, 0), PW - 1);
    y0 = min(max(y0, 0), PH - 1); x0 = min(max(x0, 0), PW - 1);
    const float* base = ps;
#define PV(yy, xx) base[(size_t)(b * PH * PW + (yy) * PW + (xx)) * KNN]
    float v00 = PV(y0, x0), v01 = PV(y0, x1), v10 = PV(y1, x0), v11 = PV(y1, x1);
#undef PV
    float top = v00 + (v01 - v00) * fx;
    float bot = v10 + (v11 - v10) * fx;
    up[idx] = top + (bot - top) * fy;
}

// ------------------------------------------------------- separable reflect-pad blur
__global__ void __launch_bounds__(256) blur_kernel(const float* __restrict__ src,
                                                   float* __restrict__ dst,
                                                   const float* __restrict__ gw, int dir) {
    __shared__ float g[KSIZE];
    if (threadIdx.x < KSIZE) g[threadIdx.x] = gw[threadIdx.x];
    __syncthreads();
    int idx = blockIdx.x * blockDim.x + threadIdx.x;
    int j = idx % OUT_HW, rest = idx / OUT_HW;
    int i = rest % OUT_HW, b = rest / OUT_HW;
    const float* base = src + (size_t)b * OUT_HW * OUT_HW;
    float acc = 0.f;
#pragma unroll
    for (int t = 0; t < KSIZE; ++t) {
        int q = (dir ? j : i) - KRAD + t;
        q = (q < 0) ? -q : ((q > OUT_HW - 1) ? 2 * (OUT_HW - 1) - q : q);
        float v = dir ? base[i * OUT_HW + q] : base[q * OUT_HW + j];
        acc += g[t] * v;
    }
    dst[idx] = acc;
}

extern "C" void kernel_launch(void* const* d_in, const int* in_sizes, int n_in,
                              void* d_out, int out_size, void* d_ws, size_t ws_size,
                              hipStream_t stream) {
    const float* emb  = (const float*)d_in[0];
    const float* memb = (const float*)d_in[1];
    const int N = in_sizes[0] / C_DIM;   // 6272
    const int M = in_sizes[1] / C_DIM;   // 100000

    float* ws  = (float*)d_ws;
    float* yn  = ws;                                   // [M]
    float* ps  = yn + M;                               // [N*9]
    float* up  = ps + (size_t)N * KNN;                 // [8*224*224]
    float* tmp = up + BATCH * OUT_HW * OUT_HW;         // [8*224*224]
    float* gw  = tmp + BATCH * OUT_HW * OUT_HW;        // [33]

    float* outMap   = (float*)d_out;                          // [8*224*224]
    float* outScore = outMap + BATCH * OUT_HW * OUT_HW;       // [8]

    gweights_kernel<<<1, 64, 0, stream>>>(gw);
    ynorm_kernel<<<512, 256, 0, stream>>>(memb, yn, M);
    cdist_topk_kernel<<<N / 16, 256, 0, stream>>>(emb, memb, yn, ps, M);
    score_kernel<<<BATCH, 256, 0, stream>>>(ps, outScore);
    const int tot = BATCH * OUT_HW * OUT_HW;
    upsample_kernel<<<tot / 256, 256, 0, stream>>>(ps, up);
    blur_kernel<<<tot / 256, 256, 0, stream>>>(up, tmp, gw, 0);
    blur_kernel<<<tot / 256, 256, 0, stream>>>(tmp, outMap, gw, 1);
}